// MultiGraphConvolutionLayers_6030134084240
// MI455X (gfx1250) — compile-verified
//
#include <hip/hip_runtime.h>
#include <math.h>

typedef float v2f __attribute__((ext_vector_type(2)));
typedef float v8f __attribute__((ext_vector_type(8)));

constexpr int Bb  = 256;   // batch
constexpr int Nn  = 128;   // nodes
constexpr int Ff  = 64;    // input features
constexpr int Ee  = 4;     // edge types
constexpr int Uu  = 128;   // hidden units (U0 == U1)
constexpr int Ff1 = Uu + Ff; // 192, layer-1 input dim

// LDS row strides (floats), padded so strided A-operand loads are bank-conflict-free
constexpr int SA0 = Ff + 4;    // 68
constexpr int SA1 = Ff1 + 4;   // 196
constexpr int SH  = Uu + 8;    // 136
constexpr int H_OFF      = Nn * SA1;            // ann region (max layer-1 size)
constexpr int LDS_FLOATS = H_OFF + Nn * SH;     // + per-edge hidden tile
// = 128*196 + 128*136 = 42496 floats = 169,984 bytes  (< 320 KB WGP LDS)

__device__ __forceinline__ v8f wmma4(v2f a, v2f b, v8f c) {
  // D(16x16,f32) = A(16x4,f32) * B(4x16,f32) + C   -- v_wmma_f32_16x16x4_f32
  return __builtin_amdgcn_wmma_f32_16x16x4_f32(
      /*neg_a=*/false, a, /*neg_b=*/false, b,
      /*c_mod=*/(short)0, c, /*reuse_a=*/false, /*reuse_b=*/false);
}

__device__ __forceinline__ float fast_tanh(float x) {
#if __has_builtin(__builtin_amdgcn_tanhf)
  return __builtin_amdgcn_tanhf(x);   // gfx1250 hardware v_tanh_f32
#else
  return tanhf(x);
#endif
}

// One GCN layer for one batch element, entirely in LDS/registers.
// Wave w computes output rows [16w, 16w+16) across all 128 columns (8 tiles).
template <int FIN, int ANNSTRIDE>
__device__ __forceinline__ void gcn_layer(
    const float* __restrict__ adjB,  // [E][N][N] for this batch element
    const float* __restrict__ W,     // [E][FIN][U]
    const float* __restrict__ bias,  // [E][U]
    const float* __restrict__ Ws,    // [FIN][U]
    const float* __restrict__ bs,    // [U]
    const float* __restrict__ ann,   // LDS, [N][ANNSTRIDE]
    float*       __restrict__ hbuf,  // LDS, [N][SH]
    float*       __restrict__ outLds,// LDS [N][SA1] (cols 0..U-1) or nullptr
    float*       __restrict__ outG)  // global [N][U] or nullptr
{
  const int lane = threadIdx.x & 31;
  const int wave = threadIdx.x >> 5;
  const int r    = lane & 15;   // column / row within half
  const int half = lane >> 4;   // K-pair selector for A/B operands
  const int m0   = wave * 16;   // this wave's output row base

  const v8f vz = {};
  v8f agg[8];
#pragma unroll
  for (int t = 0; t < 8; ++t) agg[t] = vz;

  for (int e = 0; e < Ee; ++e) {
    const float* We = W + e * FIN * Uu;

    // ---- h_e[m0:m0+16, :] = ann[m0:m0+16, :] @ W[e] + b[e] ----
    v8f hacc[8];
#pragma unroll
    for (int t = 0; t < 8; ++t) hacc[t] = vz;

    for (int k0 = 0; k0 < FIN; k0 += 4) {
      const float* ap = &ann[(m0 + r) * ANNSTRIDE + k0 + 2 * half];
      v2f a; a.x = ap[0]; a.y = ap[1];
#pragma unroll
      for (int t = 0; t < 8; ++t) {
        const float* bp = &We[(k0 + 2 * half) * Uu + 16 * t + r];
        v2f bb; bb.x = bp[0]; bb.y = bp[Uu];
        hacc[t] = wmma4(a, bb, hacc[t]);
      }
    }
    // bias + spill strip to LDS (C layout: lane<16 -> M=v, lane>=16 -> M=v+8)
#pragma unroll
    for (int t = 0; t < 8; ++t) {
      const float bv = bias[e * Uu + 16 * t + r];
#pragma unroll
      for (int v = 0; v < 8; ++v)
        hbuf[(m0 + v + 8 * half) * SH + 16 * t + r] = hacc[t][v] + bv;
    }
    __syncthreads();  // h_e fully written before anyone reads it

    // ---- agg += adj[e][m0:m0+16, :] @ h_e  (contract over neighbor n) ----
    // adjacency is streamed once per layer (twice per dispatch) -> non-temporal loads
    const float* Ae = adjB + e * Nn * Nn;
    for (int n0 = 0; n0 < Nn; n0 += 4) {
      const v2f* ap = reinterpret_cast<const v2f*>(
          &Ae[(m0 + r) * Nn + n0 + 2 * half]);   // 8B-aligned (n0%4==0, 2*half even)
      v2f a = __builtin_nontemporal_load(ap);
#pragma unroll
      for (int t = 0; t < 8; ++t) {
        const float* bp = &hbuf[(n0 + 2 * half) * SH + 16 * t + r];
        v2f bb; bb.x = bp[0]; bb.y = bp[SH];
        agg[t] = wmma4(a, bb, agg[t]);
      }
    }
    __syncthreads();  // all reads done before next edge overwrites hbuf
  }

  // ---- self loop: agg += ann @ Ws ----
  for (int k0 = 0; k0 < FIN; k0 += 4) {
    const float* ap = &ann[(m0 + r) * ANNSTRIDE + k0 + 2 * half];
    v2f a; a.x = ap[0]; a.y = ap[1];
#pragma unroll
    for (int t = 0; t < 8; ++t) {
      const float* bp = &Ws[(k0 + 2 * half) * Uu + 16 * t + r];
      v2f bb; bb.x = bp[0]; bb.y = bp[Uu];
      agg[t] = wmma4(a, bb, agg[t]);
    }
  }

  if (outLds) __syncthreads();  // everyone done reading ann before we overwrite it

#pragma unroll
  for (int t = 0; t < 8; ++t) {
    const float bv = bs[16 * t + r];
#pragma unroll
    for (int v = 0; v < 8; ++v) {
      const float val = fast_tanh(agg[t][v] + bv);
      const int row = m0 + v + 8 * half;
      const int col = 16 * t + r;
      if (outLds) outLds[row * SA1 + col] = val;                 // layer-1 input
      else        __builtin_nontemporal_store(val, &outG[row * Uu + col]);
    }
  }
}

__global__ __launch_bounds__(256) void rgcn_fused_kernel(
    const float* __restrict__ node,  // [B][N][F]
    const float* __restrict__ adj,   // [B][E][N][N]
    const float* __restrict__ W0, const float* __restrict__ b0,
    const float* __restrict__ Ws0, const float* __restrict__ bs0,
    const float* __restrict__ W1, const float* __restrict__ b1,
    const float* __restrict__ Ws1, const float* __restrict__ bs1,
    float* __restrict__ out)         // [B][N][U]
{
  extern __shared__ float lds[];
  float* annL = lds;           // layer input, [N][SA0] then [N][SA1]
  float* hbuf = lds + H_OFF;   // per-edge hidden tile, [N][SH]

  const int b   = blockIdx.x;
  const int tid = threadIdx.x;
  const float* nodeB = node + (size_t)b * Nn * Ff;
  const float* adjB  = adj  + (size_t)b * Ee * Nn * Nn;

  // stage node features -> LDS (layer-0 input, stride SA0)
  for (int i = tid; i < Nn * Ff; i += 256) {
    const int row = i / Ff, col = i % Ff;
    annL[row * SA0 + col] = nodeB[i];
  }
  __syncthreads();

  // ---- layer 0: writes tanh(h0) into annL cols [0,U) at stride SA1 ----
  gcn_layer<Ff, SA0>(adjB, W0, b0, Ws0, bs0, annL, hbuf, annL, nullptr);

  // concat: copy node features into annL cols [U, U+F) (disjoint from h0 cols)
  for (int i = tid; i < Nn * Ff; i += 256) {
    const int row = i / Ff, col = i % Ff;
    annL[row * SA1 + Uu + col] = nodeB[i];
  }
  __syncthreads();

  // ---- layer 1: writes final tanh output to global ----
  gcn_layer<Ff1, SA1>(adjB, W1, b1, Ws1, bs1, annL, hbuf, nullptr,
                      out + (size_t)b * Nn * Uu);
}

extern "C" void kernel_launch(void* const* d_in, const int* in_sizes, int n_in,
                              void* d_out, int out_size, void* d_ws, size_t ws_size,
                              hipStream_t stream) {
  const float* node = (const float*)d_in[0];
  const float* adj  = (const float*)d_in[1];
  const float* W0   = (const float*)d_in[2];
  const float* b0   = (const float*)d_in[3];
  const float* Ws0  = (const float*)d_in[4];
  const float* bs0  = (const float*)d_in[5];
  const float* W1   = (const float*)d_in[6];
  const float* b1   = (const float*)d_in[7];
  const float* Ws1  = (const float*)d_in[8];
  const float* bs1  = (const float*)d_in[9];
  float* out = (float*)d_out;

  constexpr size_t ldsBytes = (size_t)LDS_FLOATS * sizeof(float);
  hipFuncSetAttribute(reinterpret_cast<const void*>(rgcn_fused_kernel),
                      hipFuncAttributeMaxDynamicSharedMemorySize, (int)ldsBytes);

  rgcn_fused_kernel<<<dim3(Bb), dim3(256), ldsBytes, stream>>>(
      node, adj, W0, b0, Ws0, bs0, W1, b1, Ws1, bs1, out);
}